// DGDAGRNN_78872779424028
// MI455X (gfx1250) — compile-verified
//
#include <hip/hip_runtime.h>
#include <math.h>

// ---------------------------------------------------------------------------
// DGDAGRNN on gfx1250 (MI455X). f32 WMMA (V_WMMA_F32_16X16X4_F32) for all
// GEMMs. All GEMM operands are padded in workspace so fragment loads are
// unconditional v2f (b64) loads; weight tiles are staged through LDS and
// shared by 4 waves per block.
// ---------------------------------------------------------------------------

typedef __attribute__((ext_vector_type(2))) float v2f;
typedef __attribute__((ext_vector_type(8))) float v8f;

#define NN    60000
#define LL    10
#define MM    6000
#define NVT   6
#define VHS   100
#define DD    106      // VHS + NVT
#define NSV   1000
#define NRND  3
#define CHSZ  30

// padded dims
#define KP_D   108     // pad(106) to mult of 4
#define KP_V   100     // 100 already mult of 4
#define KP_C   32      // pad(30)
#define NP_V   112     // pad(100) to mult of 16
#define NP_3D  320     // pad(318)
#define NP_3V  304     // pad(300)
#define NP_C   32      // pad(30)
#define NP_O   16      // pad(3)

#define MP_E   12032   // pad(12000) to mult of 64
#define MP_M   6016    // pad(6000)
#define MP_S   1024    // pad(1000)

__device__ __forceinline__ float sigmoidf_(float x) { return 1.0f / (1.0f + expf(-x)); }

// ---------------------------------------------------------------------------
// f32 WMMA GEMM:  C[m, n] = act( A[Mpad x Kp] @ Wp[Npad x Kp]^T + bp )
// block = 128 (4 waves); wave w -> M tile (blockIdx.x*4 + w); blockIdx.y -> N
// tile. W tile (16 x Kp, contiguous in packed layout) staged via LDS.
// All fragment loads unconditional; store guarded by MstoreLimit/NoutReal.
// ---------------------------------------------------------------------------
__global__ void gemm_wmma(const float* __restrict__ A, const float* __restrict__ Wp,
                          const float* __restrict__ bp, float* __restrict__ C,
                          int Kp, int Cstride, int MstoreLimit, int NoutReal, int relu)
{
    extern __shared__ float lds[];          // 16 * Kp floats
    const int tid   = threadIdx.x;
    const int wave  = tid >> 5;
    const int lane  = tid & 31;
    const int half  = lane >> 4;
    const int lid   = lane & 15;
    const int nbase = blockIdx.y * 16;
    const int mbase = (blockIdx.x * 4 + wave) * 16;

    // cooperative load of contiguous W tile [16 x Kp]
    const float* Wtile = Wp + (size_t)nbase * Kp;
    for (int i = tid; i < 16 * Kp; i += blockDim.x) lds[i] = Wtile[i];
    __syncthreads();

    const float* Arow = A + (size_t)(mbase + lid) * Kp + 2 * half;
    const float* Brow = lds + lid * Kp + 2 * half;

    v8f acc = {};
    for (int k0 = 0; k0 < Kp; k0 += 4) {
        v2f a = *(const v2f*)(Arow + k0);
        v2f b = *(const v2f*)(Brow + k0);
        acc = __builtin_amdgcn_wmma_f32_16x16x4_f32(false, a, false, b,
                                                    (short)0, acc, false, false);
    }

    const int n = nbase + lid;
    const float bv = bp[n];
    if (n < NoutReal) {
#pragma unroll
        for (int g = 0; g < 8; ++g) {
            const int mr = mbase + g + 8 * half;
            if (mr < MstoreLimit) {
                float v = acc[g] + bv;
                if (relu) v = fmaxf(v, 0.0f);
                C[(size_t)mr * Cstride + n] = v;
            }
        }
    }
}

// ---------------------------------------------------------------------------
// Fused edge-message GEMM: msg = sigmoid(A@Wg^T + bg) * tanh(A@Wm^T)
// A: [MP_E x KP_D], Wg/Wm packed [NP_V x KP_D], msg: [MP_E x NP_V].
// ---------------------------------------------------------------------------
__global__ void msg_gemm_wmma(const float* __restrict__ A, const float* __restrict__ Wg,
                              const float* __restrict__ Wm, const float* __restrict__ bg,
                              float* __restrict__ msg, int MstoreLimit)
{
    extern __shared__ float lds[];          // 2 * 16 * KP_D floats
    float* ldsG = lds;
    float* ldsM = lds + 16 * KP_D;
    const int tid   = threadIdx.x;
    const int wave  = tid >> 5;
    const int lane  = tid & 31;
    const int half  = lane >> 4;
    const int lid   = lane & 15;
    const int nbase = blockIdx.y * 16;
    const int mbase = (blockIdx.x * 4 + wave) * 16;

    const float* Gt = Wg + (size_t)nbase * KP_D;
    const float* Mt = Wm + (size_t)nbase * KP_D;
    for (int i = tid; i < 16 * KP_D; i += blockDim.x) { ldsG[i] = Gt[i]; ldsM[i] = Mt[i]; }
    __syncthreads();

    const float* Arow  = A + (size_t)(mbase + lid) * KP_D + 2 * half;
    const float* BGrow = ldsG + lid * KP_D + 2 * half;
    const float* BMrow = ldsM + lid * KP_D + 2 * half;

    v8f accG = {};
    v8f accM = {};
    for (int k0 = 0; k0 < KP_D; k0 += 4) {
        v2f a  = *(const v2f*)(Arow + k0);
        v2f bG = *(const v2f*)(BGrow + k0);
        v2f bM = *(const v2f*)(BMrow + k0);
        accG = __builtin_amdgcn_wmma_f32_16x16x4_f32(false, a, false, bG,
                                                     (short)0, accG, false, false);
        accM = __builtin_amdgcn_wmma_f32_16x16x4_f32(false, a, false, bM,
                                                     (short)0, accM, false, false);
    }

    const int n = nbase + lid;
    const float bgv = bg[n];
#pragma unroll
    for (int g = 0; g < 8; ++g) {
        const int mr = mbase + g + 8 * half;
        if (mr < MstoreLimit) {
            msg[(size_t)mr * NP_V + n] = sigmoidf_(accG[g] + bgv) * tanhf(accM[g]);
        }
    }
}

// ---------------------------------------------------------------------------
// Packing / elementwise / gather kernels
// ---------------------------------------------------------------------------

// Wp[n, k] = (n<N && k<K) ? W[n, k] : 0
__global__ void pack_w_kernel(const float* __restrict__ W, float* __restrict__ out,
                              int N, int K, int Npad, int Kp)
{
    int t = blockIdx.x * blockDim.x + threadIdx.x;
    if (t >= Npad * Kp) return;
    int n = t / Kp, k = t % Kp;
    out[t] = (n < N && k < K) ? W[n * K + k] : 0.0f;
}

__global__ void pack_b_kernel(const float* __restrict__ b, float* __restrict__ out,
                              int N, int Npad)
{
    int t = blockIdx.x * blockDim.x + threadIdx.x;
    if (t >= Npad) return;
    out[t] = (t < N) ? b[t] : 0.0f;
}

__global__ void zero_kernel(float* __restrict__ p, int n)
{
    int t = blockIdx.x * blockDim.x + threadIdx.x;
    if (t < n) p[t] = 0.0f;
}

// h[i, d] = w_init[d]
__global__ void init_h_kernel(float* __restrict__ h, const float* __restrict__ w_init)
{
    int t = blockIdx.x * blockDim.x + threadIdx.x;
    if (t >= NN * VHS) return;
    h[t] = w_init[t % VHS];
}

// out[r, :] = concat(h[node], x[node]) padded to KP_D; zero pad rows/cols.
// node = idx ? idx[r] : base + r
__global__ void build_cat_kernel(const float* __restrict__ h, const float* __restrict__ x,
                                 const int* __restrict__ idx, int base,
                                 float* __restrict__ out, int rowsReal, int rowsPad)
{
    int t = blockIdx.x * blockDim.x + threadIdx.x;
    if (t >= rowsPad * KP_D) return;
    int r = t / KP_D, d = t % KP_D;
    float v = 0.0f;
    if (r < rowsReal && d < DD) {
        int node = idx ? idx[r] : (base + r);
        v = (d < VHS) ? h[node * VHS + d] : x[node * NVT + (d - VHS)];
    }
    out[t] = v;
}

// ps[j, d] = (msg[2j, d] + msg[2j+1, d]) / 2 ; zero pad rows (j >= MM).
__global__ void pair_avg_kernel(const float* __restrict__ msg, float* __restrict__ ps)
{
    int t = blockIdx.x * blockDim.x + threadIdx.x;
    if (t >= MP_M * KP_V) return;
    int j = t / KP_V, d = t % KP_V;
    float v = 0.0f;
    if (j < MM) v = 0.5f * (msg[(size_t)(2 * j) * NP_V + d] + msg[(size_t)(2 * j + 1) * NP_V + d]);
    ps[t] = v;
}

// out[r, 0:100] = h[node] ; zero pad rows. node = bcast ? idx[0] : idx[r]
__global__ void gather_rows_kernel(const float* __restrict__ h, const int* __restrict__ idx,
                                   int bcast, float* __restrict__ out, int rowsReal, int rowsPad)
{
    int t = blockIdx.x * blockDim.x + threadIdx.x;
    if (t >= rowsPad * VHS) return;
    int r = t / VHS, d = t % VHS;
    float v = 0.0f;
    if (r < rowsReal) {
        int node = bcast ? idx[0] : idx[r];
        v = h[node * VHS + d];
    }
    out[t] = v;
}

// GRU gate combine: gi/gh rows have stride gstride (3 chunks at 0,H,2H),
// hid rows have stride hstride. Writes out[row, d] for d < out_limit.
__global__ void gru_combine_kernel(const float* __restrict__ gi, int gstride,
                                   const float* __restrict__ gh,
                                   const float* __restrict__ hid, int hstride, int H,
                                   float* __restrict__ out, const int* __restrict__ out_idx,
                                   int out_stride, int out_limit, int rows)
{
    int t = blockIdx.x * blockDim.x + threadIdx.x;
    if (t >= rows * H) return;
    int r = t / H, d = t % H;
    const float* girow = gi + (size_t)r * gstride;
    const float* ghrow = gh + (size_t)r * gstride;
    float ir = girow[d],         hr = ghrow[d];
    float iz = girow[H + d],     hz = ghrow[H + d];
    float in = girow[2 * H + d], hn = ghrow[2 * H + d];
    float rg = sigmoidf_(ir + hr);
    float z  = sigmoidf_(iz + hz);
    float nv = tanhf(in + rg * hn);
    float v  = (1.0f - z) * nv + z * hid[(size_t)r * hstride + d];
    if (d < out_limit) {
        int row = out_idx ? out_idx[r] : r;
        out[(size_t)row * out_stride + d] = v;
    }
}

// ---------------------------------------------------------------------------
// Host-side launch
// ---------------------------------------------------------------------------
extern "C" void kernel_launch(void* const* d_in, const int* in_sizes, int n_in,
                              void* d_out, int out_size, void* d_ws, size_t ws_size,
                              hipStream_t stream)
{
    (void)in_sizes; (void)n_in; (void)out_size; (void)ws_size;

    const float* x        = (const float*)d_in[0];
    const float* w_init   = (const float*)d_in[1];
    const float* gate_w   = (const float*)d_in[2];
    const float* gate_b   = (const float*)d_in[3];
    const float* map_w    = (const float*)d_in[4];
    const float* grue_wih = (const float*)d_in[5];
    const float* grue_whh = (const float*)d_in[6];
    const float* grue_bih = (const float*)d_in[7];
    const float* grue_bhh = (const float*)d_in[8];
    const float* gli_wih  = (const float*)d_in[9];
    const float* gli_whh  = (const float*)d_in[10];
    const float* gli_bih  = (const float*)d_in[11];
    const float* gli_bhh  = (const float*)d_in[12];
    const float* gout_wih = (const float*)d_in[13];
    const float* gout_whh = (const float*)d_in[14];
    const float* gout_bih = (const float*)d_in[15];
    const float* gout_bhh = (const float*)d_in[16];
    const float* mlp_w1   = (const float*)d_in[17];
    const float* mlp_b1   = (const float*)d_in[18];
    const float* mlp_w2   = (const float*)d_in[19];
    const float* mlp_b2   = (const float*)d_in[20];
    const int* edge_src   = (const int*)d_in[21];
    const int* sv_node    = (const int*)d_in[23];
    const int* li_node    = (const int*)d_in[24];
    const int* out_node   = (const int*)d_in[25];

    // ---------------- workspace layout (float offsets) ----------------
    float* ws = (float*)d_ws;
    float* p  = ws;
    float* h       = p; p += (size_t)NN * VHS;        // 6,000,000
    float* Abuf    = p; p += (size_t)MP_E * KP_D;     // 1,299,456
    float* msg     = p; p += (size_t)MP_E * NP_V;     // 1,347,584
    float* ps      = p; p += (size_t)MP_M * KP_V;     //   601,600
    float* gi      = p; p += (size_t)MM * NP_3D;      // 1,920,000
    float* gh      = p; p += (size_t)MM * NP_3D;      // 1,920,000
    float* li_buf  = p; p += (size_t)MP_S * VHS;
    float* svprev  = p; p += (size_t)MP_S * VHS;
    float* outh    = p; p += (size_t)MP_S * VHS;
    float* gi2     = p; p += (size_t)MP_S * NP_3V;
    float* gh2     = p; p += (size_t)MP_S * NP_3V;
    float* afterli = p; p += (size_t)MP_S * VHS;
    float* svfeat  = p; p += (size_t)MP_S * VHS;
    float* hid1    = p; p += (size_t)MP_S * NP_C;
    // packed weights
    float* gate_wp   = p; p += NP_V * KP_D;
    float* map_wp    = p; p += NP_V * KP_D;
    float* gate_bp   = p; p += NP_V;
    float* grue_wihp = p; p += NP_3D * KP_V;
    float* grue_whhp = p; p += NP_3D * KP_D;
    float* grue_bihp = p; p += NP_3D;
    float* grue_bhhp = p; p += NP_3D;
    float* gli_wihp  = p; p += NP_3V * KP_V;
    float* gli_whhp  = p; p += NP_3V * KP_V;
    float* gli_bihp  = p; p += NP_3V;
    float* gli_bhhp  = p; p += NP_3V;
    float* gout_wihp = p; p += NP_3V * KP_V;
    float* gout_whhp = p; p += NP_3V * KP_V;
    float* gout_bihp = p; p += NP_3V;
    float* gout_bhhp = p; p += NP_3V;
    float* mlp_w1p   = p; p += NP_C * KP_V;
    float* mlp_b1p   = p; p += NP_C;
    float* mlp_w2p   = p; p += NP_O * KP_C;
    float* mlp_b2p   = p; p += NP_O;

    const int TB = 256;
    auto blks = [](long n, int tb) { return (int)((n + tb - 1) / tb); };
#define PACKW(src, dst, N, K, Np, Kp) \
    pack_w_kernel<<<blks((long)(Np) * (Kp), TB), TB, 0, stream>>>(src, dst, N, K, Np, Kp)
#define PACKB(src, dst, N, Np) \
    pack_b_kernel<<<blks((long)(Np), TB), TB, 0, stream>>>(src, dst, N, Np)

    // ---------------- one-time packing / init ----------------
    PACKW(gate_w,   gate_wp,   VHS,     DD,   NP_V,  KP_D);
    PACKW(map_w,    map_wp,    VHS,     DD,   NP_V,  KP_D);
    PACKB(gate_b,   gate_bp,   VHS,     NP_V);
    PACKW(grue_wih, grue_wihp, 3 * DD,  VHS,  NP_3D, KP_V);
    PACKW(grue_whh, grue_whhp, 3 * DD,  DD,   NP_3D, KP_D);
    PACKB(grue_bih, grue_bihp, 3 * DD,  NP_3D);
    PACKB(grue_bhh, grue_bhhp, 3 * DD,  NP_3D);
    PACKW(gli_wih,  gli_wihp,  3 * VHS, VHS,  NP_3V, KP_V);
    PACKW(gli_whh,  gli_whhp,  3 * VHS, VHS,  NP_3V, KP_V);
    PACKB(gli_bih,  gli_bihp,  3 * VHS, NP_3V);
    PACKB(gli_bhh,  gli_bhhp,  3 * VHS, NP_3V);
    PACKW(gout_wih, gout_wihp, 3 * VHS, VHS,  NP_3V, KP_V);
    PACKW(gout_whh, gout_whhp, 3 * VHS, VHS,  NP_3V, KP_V);
    PACKB(gout_bih, gout_bihp, 3 * VHS, NP_3V);
    PACKB(gout_bhh, gout_bhhp, 3 * VHS, NP_3V);
    PACKW(mlp_w1,   mlp_w1p,   CHSZ,    VHS,  NP_C,  KP_V);
    PACKB(mlp_b1,   mlp_b1p,   CHSZ,    NP_C);
    PACKW(mlp_w2,   mlp_w2p,   3,       CHSZ, NP_O,  KP_C);
    PACKB(mlp_b2,   mlp_b2p,   3,       NP_O);

    // zero pad rows of afterli (rows NSV..MP_S-1 must be 0 as GEMM A input)
    zero_kernel<<<blks((long)(MP_S - NSV) * VHS, TB), TB, 0, stream>>>(
        afterli + (size_t)NSV * VHS, (MP_S - NSV) * VHS);

    // h = broadcast(w_init)
    init_h_kernel<<<blks((long)NN * VHS, TB), TB, 0, stream>>>(h, w_init);

    const int smemD = 16 * KP_D * sizeof(float);
    const int smemV = 16 * KP_V * sizeof(float);
    const int smemC = 16 * KP_C * sizeof(float);

    for (int r = 0; r < NRND; ++r) {
        for (int l = 1; l < LL; ++l) {
            const int e0 = (l - 1) * 2 * MM;

            // hsrc = concat(h[src], x[src]) : [MP_E x KP_D], zero padded
            build_cat_kernel<<<blks((long)MP_E * KP_D, TB), TB, 0, stream>>>(
                h, x, edge_src + e0, 0, Abuf, 2 * MM, MP_E);

            // msg = sigmoid(hsrc@gate_w^T + b) * tanh(hsrc@map_w^T)
            msg_gemm_wmma<<<dim3(MP_E / 64, NP_V / 16), 128, 2 * smemD, stream>>>(
                Abuf, gate_wp, map_wp, gate_bp, msg, 2 * MM);

            // ps_h[j] = (msg[2j] + msg[2j+1]) / 2 (zero pad rows)
            pair_avg_kernel<<<blks((long)MP_M * KP_V, TB), TB, 0, stream>>>(msg, ps);

            // inp = concat(h[l*M + j], x[l*M + j]) : [MP_M x KP_D]
            build_cat_kernel<<<blks((long)MP_M * KP_D, TB), TB, 0, stream>>>(
                h, x, nullptr, l * MM, Abuf, MM, MP_M);

            // gi = ps @ grue_wih^T + bih   (stride NP_3D)
            gemm_wmma<<<dim3(MP_M / 64, NP_3D / 16), 128, smemV, stream>>>(
                ps, grue_wihp, grue_bihp, gi, KP_V, NP_3D, MM, NP_3D, 0);
            // gh = inp @ grue_whh^T + bhh  (stride NP_3D)
            gemm_wmma<<<dim3(MP_M / 64, NP_3D / 16), 128, smemD, stream>>>(
                Abuf, grue_whhp, grue_bhhp, gh, KP_D, NP_3D, MM, NP_3D, 0);

            // h[l*M:(l+1)*M, :] = gru(...)[:, :100]
            gru_combine_kernel<<<blks((long)MM * DD, TB), TB, 0, stream>>>(
                gi, NP_3D, gh, Abuf, KP_D, DD,
                h + (size_t)l * MM * VHS, nullptr, VHS, VHS, MM);
        }

        // sv / li / out gathers (zero padded rows)
        gather_rows_kernel<<<blks((long)MP_S * VHS, TB), TB, 0, stream>>>(h, li_node, 0, li_buf, NSV, MP_S);
        gather_rows_kernel<<<blks((long)MP_S * VHS, TB), TB, 0, stream>>>(h, sv_node, 0, svprev, NSV, MP_S);
        gather_rows_kernel<<<blks((long)MP_S * VHS, TB), TB, 0, stream>>>(h, out_node, 1, outh, NSV, MP_S);

        // after_li = gru(li, sv_prev, gli)
        gemm_wmma<<<dim3(MP_S / 64, NP_3V / 16), 128, smemV, stream>>>(
            li_buf, gli_wihp, gli_bihp, gi2, KP_V, NP_3V, MP_S, NP_3V, 0);
        gemm_wmma<<<dim3(MP_S / 64, NP_3V / 16), 128, smemV, stream>>>(
            svprev, gli_whhp, gli_bhhp, gh2, KP_V, NP_3V, MP_S, NP_3V, 0);
        gru_combine_kernel<<<blks((long)NSV * VHS, TB), TB, 0, stream>>>(
            gi2, NP_3V, gh2, svprev, VHS, VHS, afterli, nullptr, VHS, VHS, NSV);

        // h[sv_node] = gru(out_h, after_li, gout)
        gemm_wmma<<<dim3(MP_S / 64, NP_3V / 16), 128, smemV, stream>>>(
            outh, gout_wihp, gout_bihp, gi2, KP_V, NP_3V, MP_S, NP_3V, 0);
        gemm_wmma<<<dim3(MP_S / 64, NP_3V / 16), 128, smemV, stream>>>(
            afterli, gout_whhp, gout_bhhp, gh2, KP_V, NP_3V, MP_S, NP_3V, 0);
        gru_combine_kernel<<<blks((long)NSV * VHS, TB), TB, 0, stream>>>(
            gi2, NP_3V, gh2, afterli, VHS, VHS, h, sv_node, VHS, VHS, NSV);
    }

    // Final MLP: relu(relu(h[sv] @ w1^T + b1) @ w2^T + b2) -> d_out (1000 x 3)
    gather_rows_kernel<<<blks((long)MP_S * VHS, TB), TB, 0, stream>>>(h, sv_node, 0, svfeat, NSV, MP_S);
    gemm_wmma<<<dim3(MP_S / 64, NP_C / 16), 128, smemV, stream>>>(
        svfeat, mlp_w1p, mlp_b1p, hid1, KP_V, NP_C, MP_S, NP_C, 1);
    gemm_wmma<<<dim3(MP_S / 64, NP_O / 16), 128, smemC, stream>>>(
        hid1, mlp_w2p, mlp_b2p, (float*)d_out, KP_C, 3, NSV, 3, 1);
}